// Model_routine_1236950581278
// MI455X (gfx1250) — compile-verified
//
#include <hip/hip_runtime.h>
#include <cmath>
#include <stdint.h>

// ---------------------------------------------------------------------------
// MI455X (gfx1250) implementation. wave32. fp32 WMMA (V_WMMA_F32_16X16X4_F32)
// for every GEMM: workload is ~11.5 GFLOP vs ~150-200MB traffic -> memory/
// latency bound; fp32 keeps validation-exact precision while driving the
// matrix pipe. GEMM: 2x4 tiles/wave (6 b64 loads : 8 WMMAs per k-step),
// float2-unit 32-bit offsets (SGPR-base + VGPR-offset b64 loads), software-
// pipelined, branch-free, uniform EXEC.
// ---------------------------------------------------------------------------

typedef __attribute__((ext_vector_type(2))) float v2f;
typedef __attribute__((ext_vector_type(8))) float v8f;

#define BSZ   128
#define LSEQ  176
#define ROWS  (BSZ * LSEQ)      // 22528, divisible by 128

// ---------------------------------------------------------------------------
// Generic fp32 WMMA GEMM:  C[M,N] = A[M,K] @ W[N,K]^T (+bias) (+activation)
//   Each wave computes a 32 x (16*NT) strip (MT=2 m-tiles, NT n-tiles).
//   Launch contract: gridDim.x*4 waves cover M/32 exactly; gridDim.y*NT ==
//   ceil(N/16); lda and K must be even (true for every layer here).
//   RAGGED=true only when N % 16 != 0 (x_proj N=24, fc3 N=15): B rows are
//   pointer-clamped + zero-masked (branch-free, EXEC stays all-ones).
//   act: 0=none, 1=sigmoid, 2=softplus
// Fragment layout per CDNA5 ISA 7.12.2 (32-bit 16x4 A, 4x16 B, 16x16 C/D).
// ---------------------------------------------------------------------------
template<int NT, bool RAGGED>
__global__ void gemm_wmma_kernel(const float* __restrict__ A, int lda,
                                 const float* __restrict__ W, int K,
                                 const float* __restrict__ bias,
                                 float* __restrict__ C, int ldc,
                                 int M, int N, int act)
{
    const int wave  = threadIdx.x >> 5;
    const int lane  = threadIdx.x & 31;
    const int mPair = blockIdx.x * 4 + wave;     // 2 m-tiles per wave
    const int m0    = mPair * 32;
    if (m0 >= M) return;                         // uniform per wave

    const int half = lane >> 4;                  // 0: lanes 0-15, 1: lanes 16-31
    const int idx  = lane & 15;
    const int nt0  = blockIdx.y * NT;

    const v8f zero = {0.f,0.f,0.f,0.f,0.f,0.f,0.f,0.f};
    v8f acc[2][NT];
#pragma unroll
    for (int i = 0; i < 2; ++i)
#pragma unroll
        for (int j = 0; j < NT; ++j) acc[i][j] = zero;

    // Index in float2 units: one b64 load per fragment, SGPR base + 32-bit
    // VGPR offset. lda/K even for all layers; all buffers 8B-aligned.
    const v2f* __restrict__ A2 = (const v2f*)A;
    const v2f* __restrict__ W2 = (const v2f*)W;
    const uint32_t lda2 = (uint32_t)lda >> 1;
    const uint32_t K2   = (uint32_t)K >> 1;

    // A fragment i: lane holds row m0+16i+idx, K-slots 2*half+{0,1}
    uint32_t aoff[2];
#pragma unroll
    for (int i = 0; i < 2; ++i)
        aoff[i] = (uint32_t)(m0 + 16 * i + idx) * lda2 + half;

    // B fragment j: lane holds W row (nt0+j)*16+idx, K-slots 2*half+{0,1}
    uint32_t woff[NT];
    float bmask[NT];
#pragma unroll
    for (int j = 0; j < NT; ++j) {
        int n = (nt0 + j) * 16 + idx;
        if (RAGGED) {
            bmask[j] = (n < N) ? 1.f : 0.f;
            if (n > N - 1) n = N - 1;            // clamp: keep loads in-bounds
        }
        woff[j] = (uint32_t)n * K2 + half;
    }

    auto load_frag = [&](uint32_t k2, v2f (&a)[2], v2f (&b)[NT]) {
#pragma unroll
        for (int i = 0; i < 2; ++i)
            a[i] = A2[aoff[i] + k2];
#pragma unroll
        for (int j = 0; j < NT; ++j) {
            b[j] = W2[woff[j] + k2];
            if (RAGGED) { b[j].x *= bmask[j]; b[j].y *= bmask[j]; }
        }
    };

    // Software pipeline: prefetch the next k-step's fragments before this
    // step's 8-WMMA group, so loads overlap WMMA. Final step peeled (no OOB
    // prefetch). k2 advances 2 float2 = 4 K-slots per step.
    v2f a_cur[2], b_cur[NT];
    load_frag(0, a_cur, b_cur);
    uint32_t k2 = 0;
#pragma unroll 2
    for (; k2 + 2 < K2; k2 += 2) {
        v2f a_nxt[2], b_nxt[NT];
        load_frag(k2 + 2, a_nxt, b_nxt);
#pragma unroll
        for (int i = 0; i < 2; ++i)
#pragma unroll
            for (int j = 0; j < NT; ++j)
                acc[i][j] = __builtin_amdgcn_wmma_f32_16x16x4_f32(
                    false, a_cur[i], false, b_cur[j], (short)0, acc[i][j], false, false);
#pragma unroll
        for (int i = 0; i < 2; ++i) a_cur[i] = a_nxt[i];
#pragma unroll
        for (int j = 0; j < NT; ++j) b_cur[j] = b_nxt[j];
    }
#pragma unroll
    for (int i = 0; i < 2; ++i)
#pragma unroll
        for (int j = 0; j < NT; ++j)
            acc[i][j] = __builtin_amdgcn_wmma_f32_16x16x4_f32(
                false, a_cur[i], false, b_cur[j], (short)0, acc[i][j], false, false);

    // Epilogue: C/D layout -> row = m0 + 16i + r + 8*half, col = nt*16 + idx
#pragma unroll
    for (int j = 0; j < NT; ++j) {
        const int col = (nt0 + j) * 16 + idx;
        if (RAGGED && col >= N) continue;
        const float bv = bias ? bias[col] : 0.f;
#pragma unroll
        for (int i = 0; i < 2; ++i) {
#pragma unroll
            for (int r = 0; r < 8; ++r) {
                const int row = m0 + 16 * i + r + 8 * half;
                float v = acc[i][j][r] + bv;
                if (act == 1)       v = 1.f / (1.f + expf(-v));                  // sigmoid
                else if (act == 2)  v = fmaxf(v, 0.f) + log1pf(expf(-fabsf(v))); // softplus
                C[(size_t)row * ldc + col] = v;
            }
        }
    }
}

// ---------------------------------------------------------------------------
// Fused bn1(eval) + leaky_relu(0.01) + LayerNorm(256), in-place.
// One wave32 per row (8 elems/lane), shfl_xor tree reduction.
// ---------------------------------------------------------------------------
__global__ void bn_leaky_ln_kernel(float* __restrict__ buf,
                                   const float* __restrict__ bn_g,
                                   const float* __restrict__ bn_b,
                                   const float* __restrict__ ln_g,
                                   const float* __restrict__ ln_b)
{
    const int wave = threadIdx.x >> 5;
    const int lane = threadIdx.x & 31;
    const int row  = blockIdx.x * 8 + wave;
    if (row >= ROWS) return;
    const int t = row % LSEQ;
    const float sc = bn_g[t] * rsqrtf(1.f + 1e-5f);
    const float sh = bn_b[t];

    float* p = buf + (size_t)row * 256;
    float v[8];
    float s = 0.f, sq = 0.f;
#pragma unroll
    for (int i = 0; i < 8; ++i) {
        float x = p[lane + 32 * i];
        x = x * sc + sh;
        x = (x > 0.f) ? x : 0.01f * x;           // leaky_relu
        v[i] = x; s += x; sq += x * x;
    }
#pragma unroll
    for (int m = 16; m >= 1; m >>= 1) {
        s  += __shfl_xor(s,  m, 32);
        sq += __shfl_xor(sq, m, 32);
    }
    const float mu  = s * (1.f / 256.f);
    const float var = sq * (1.f / 256.f) - mu * mu;
    const float inv = rsqrtf(var + 1e-5f);
#pragma unroll
    for (int i = 0; i < 8; ++i) {
        const int c = lane + 32 * i;
        p[c] = (v[i] - mu) * inv * ln_g[c] + ln_b[c];
    }
}

// ---------------------------------------------------------------------------
// Depthwise causal conv (k=2, zero left-pad) + SiLU.
// u = xz[:, :, 0:256] (stride 512).
// ---------------------------------------------------------------------------
__global__ void conv_silu_kernel(const float* __restrict__ xz,
                                 const float* __restrict__ conv_w,  // (256,1,2)
                                 const float* __restrict__ conv_b,
                                 float* __restrict__ xc)
{
    const int gid = blockIdx.x * blockDim.x + threadIdx.x;
    if (gid >= ROWS * 256) return;
    const int row = gid >> 8;
    const int d   = gid & 255;
    const int t   = row % LSEQ;
    const float ucur  = xz[(size_t)row * 512 + d];
    const float uprev = (t > 0) ? xz[(size_t)(row - 1) * 512 + d] : 0.f;
    const float v = uprev * conv_w[2 * d] + ucur * conv_w[2 * d + 1] + conv_b[d];
    xc[gid] = v / (1.f + expf(-v));              // silu
}

// ---------------------------------------------------------------------------
// Selective-scan: one thread per (b, d) channel, D_STATE=4 state in regs.
// Fuses y = (scan + Dp*xc) * silu(z). Bm/Cm live in dbl[:,16:20]/[:,20:24];
// broadcast loads across the 256 d-threads hit L0/L2.
// ---------------------------------------------------------------------------
__global__ void mamba_scan_kernel(const float* __restrict__ dt,
                                  const float* __restrict__ xc,
                                  const float* __restrict__ dbl,   // ROWS x 24
                                  const float* __restrict__ xz,    // z at cols 256..511
                                  const float* __restrict__ A_log, // (256,4)
                                  const float* __restrict__ Dp,    // (256,)
                                  float* __restrict__ yact)
{
    const int d = threadIdx.x;                   // 0..255
    const int b = blockIdx.x;                    // 0..127
    float An[4];
#pragma unroll
    for (int n = 0; n < 4; ++n) An[n] = -expf(A_log[d * 4 + n]);
    const float Dpd = Dp[d];
    float h[4] = {0.f, 0.f, 0.f, 0.f};
    const size_t r0 = (size_t)b * LSEQ;

    for (int t = 0; t < LSEQ; ++t) {
        const size_t row = r0 + t;
        const float dtv = dt[row * 256 + d];
        const float xcv = xc[row * 256 + d];
        const float* dr = dbl + row * 24;
        float y = 0.f;
#pragma unroll
        for (int n = 0; n < 4; ++n) {
            const float dA = expf(dtv * An[n]);
            h[n] = h[n] * dA + dtv * dr[16 + n] * xcv;   // dBu
            y += h[n] * dr[20 + n];                      // * Cm
        }
        const float z = xz[row * 512 + 256 + d];
        yact[row * 256 + d] = (y + Dpd * xcv) * (z / (1.f + expf(-z)));
    }
}

// ---------------------------------------------------------------------------
// bn5 (eval), elementwise in-place on ROWS x 256.
// ---------------------------------------------------------------------------
__global__ void bn5_kernel(float* __restrict__ buf,
                           const float* __restrict__ g,
                           const float* __restrict__ b)
{
    const int gid = blockIdx.x * blockDim.x + threadIdx.x;
    if (gid >= ROWS * 256) return;
    const int t = (gid >> 8) % LSEQ;
    buf[gid] = buf[gid] * (g[t] * rsqrtf(1.f + 1e-5f)) + b[t];
}

// ---------------------------------------------------------------------------
extern "C" void kernel_launch(void* const* d_in, const int* in_sizes, int n_in,
                              void* d_out, int out_size, void* d_ws, size_t ws_size,
                              hipStream_t stream)
{
    const float* x         = (const float*)d_in[0];
    const float* lin1_w    = (const float*)d_in[1];
    const float* lin1_b    = (const float*)d_in[2];
    const float* bn1_g     = (const float*)d_in[3];
    const float* bn1_b     = (const float*)d_in[4];
    const float* ln_g      = (const float*)d_in[5];
    const float* ln_b      = (const float*)d_in[6];
    const float* in_proj_w = (const float*)d_in[7];
    const float* conv_w    = (const float*)d_in[8];
    const float* conv_b    = (const float*)d_in[9];
    const float* x_proj_w  = (const float*)d_in[10];
    const float* dt_proj_w = (const float*)d_in[11];
    const float* dt_proj_b = (const float*)d_in[12];
    const float* A_log     = (const float*)d_in[13];
    const float* Dp        = (const float*)d_in[14];
    const float* out_proj_w= (const float*)d_in[15];
    const float* bn5_g     = (const float*)d_in[16];
    const float* bn5_b     = (const float*)d_in[17];
    const float* l5a_w     = (const float*)d_in[18];
    const float* l5a_b     = (const float*)d_in[19];
    const float* l5b_w     = (const float*)d_in[20];
    const float* l5b_b     = (const float*)d_in[21];
    const float* l5c_w     = (const float*)d_in[22];
    const float* l5c_b     = (const float*)d_in[23];
    const float* fc3_w     = (const float*)d_in[24];
    const float* fc3_b     = (const float*)d_in[25];
    float* out = (float*)d_out;

    // Workspace layout (floats); buffers reused across phases.
    float* ws = (float*)d_ws;
    const size_t S = (size_t)ROWS * 256;         // 5,767,168 floats
    float* W0 = ws;                              //  S : h1/xln -> dt -> l5a -> l5c
    float* W1 = ws + S;                          // 2S : xz
    float* W2 = ws + 3 * S;                      //  S : xc -> y2
    float* W3 = ws + 4 * S;                      //  ROWS*24 : dbl
    float* W4 = ws + 4 * S + (size_t)ROWS * 24;  //  S : yact -> l5b

    const dim3 gblk(128);                        // 4 waves / block, 32 rows/wave
    const int MB = ROWS / 128;                   // 176 GEMM blocks (m-dim)
    const int EW = (ROWS * 256 + 255) / 256;     // elementwise blocks

    // 1) lin1: (ROWS,20)@(20,256)+b -> W0                     [16 tiles = 4x4]
    gemm_wmma_kernel<4, false><<<dim3(MB, 4), gblk, 0, stream>>>(
        x, 20, lin1_w, 20, lin1_b, W0, 256, ROWS, 256, 0);
    // 2) bn1 + leaky + LayerNorm, in-place on W0
    bn_leaky_ln_kernel<<<dim3(ROWS / 8), dim3(256), 0, stream>>>(
        W0, bn1_g, bn1_b, ln_g, ln_b);
    // 3) in_proj: (ROWS,256)@(256,512) -> W1 (xz)             [32 tiles = 8x4]
    gemm_wmma_kernel<4, false><<<dim3(MB, 8), gblk, 0, stream>>>(
        W0, 256, in_proj_w, 256, nullptr, W1, 512, ROWS, 512, 0);
    // 4) conv(k=2) + silu -> W2 (xc)
    conv_silu_kernel<<<dim3(EW), dim3(256), 0, stream>>>(W1, conv_w, conv_b, W2);
    // 5) x_proj: (ROWS,256)@(256,24) -> W3 (dbl)              [2 ragged tiles]
    gemm_wmma_kernel<1, true><<<dim3(MB, 2), gblk, 0, stream>>>(
        W2, 256, x_proj_w, 256, nullptr, W3, 24, ROWS, 24, 0);
    // 6) dt = softplus(dtr @ dt_proj^T + b) -> W0 (K=16 slice of dbl, lda=24)
    gemm_wmma_kernel<4, false><<<dim3(MB, 4), gblk, 0, stream>>>(
        W3, 24, dt_proj_w, 16, dt_proj_b, W0, 256, ROWS, 256, 2);
    // 7) selective scan + *silu(z) -> W4 (yact)
    mamba_scan_kernel<<<dim3(BSZ), dim3(256), 0, stream>>>(
        W0, W2, W3, W1, A_log, Dp, W4);
    // 8) out_proj -> W2 (y2)                                  [16 tiles = 4x4]
    gemm_wmma_kernel<4, false><<<dim3(MB, 4), gblk, 0, stream>>>(
        W4, 256, out_proj_w, 256, nullptr, W2, 256, ROWS, 256, 0);
    // 9) bn5 in-place
    bn5_kernel<<<dim3(EW), dim3(256), 0, stream>>>(W2, bn5_g, bn5_b);
    // 10) l5a -> W0 [8 tiles = 2x4], l5b -> W4 [4 tiles = 1x4], l5c -> W0 [1 tile]
    gemm_wmma_kernel<4, false><<<dim3(MB, 2), gblk, 0, stream>>>(
        W2, 256, l5a_w, 256, l5a_b, W0, 128, ROWS, 128, 0);
    gemm_wmma_kernel<4, false><<<dim3(MB, 1), gblk, 0, stream>>>(
        W0, 128, l5b_w, 128, l5b_b, W4, 64, ROWS, 64, 0);
    gemm_wmma_kernel<1, false><<<dim3(MB, 1), gblk, 0, stream>>>(
        W4, 64, l5c_w, 64, l5c_b, W0, 16, ROWS, 16, 0);
    // 11) fc3 + sigmoid: W0 viewed as (128, 2816) -> out (128,15) [1 ragged tile]
    gemm_wmma_kernel<1, true><<<dim3(1, 1), gblk, 0, stream>>>(
        W0, 2816, fc3_w, 2816, fc3_b, out, 15, BSZ, 15, 1);
}